// CBOWScratch_81827716924176
// MI455X (gfx1250) — compile-verified
//
#include <hip/hip_runtime.h>

// CBOW negative-sampling loss for MI455X (gfx1250, wave32).
// Memory-bound gather problem (~176 MB @ 23.3 TB/s ≈ 7.6 us floor; both
// embedding tables total 102 MB < 192 MB L2, so gathers are L2-resident).
// Scores run on V_WMMA_F32_16X16X4_F32 (full f32 = reference precision):
// per wave, a 16x16 GEMM tile D = mean_tile x vec_tile^T whose DIAGONAL is
// the 16 per-row scores for column j (1 target + 10 negatives).
//
// CDNA5 specifics used:
//  - v_wmma_f32_16x16x4_f32 chains (two independent accumulators for ILP)
//  - per-wave LDS staging with explicit `s_wait_dscnt 0` instead of block
//    barriers (LDS ops are in-order within a wave; waves own disjoint LDS)
//  - A-fragments (context-mean tile) hoisted into VGPRs across all 11 columns

#define VOCABN 100000
#define DIM    128
#define BN     16384
#define CWIN   10
#define KNEG   10
#define EPSF   1e-9f

#define ROWS   16   // b-rows per wave (= WMMA tile M/N)
#define WAVES  2    // waves per workgroup
#define LPAD   132  // padded LDS row stride in floats (bank-conflict free)

typedef __attribute__((ext_vector_type(2))) float v2f;
typedef __attribute__((ext_vector_type(8))) float v8f;

__device__ __forceinline__ void wave_lds_fence() {
    // Within-wave cross-lane LDS visibility: LDS ops are in-order per wave,
    // so draining DScnt is sufficient; "memory" clobber stops reordering.
    asm volatile("s_wait_dscnt 0" ::: "memory");
}

__global__ void cbow_zero_kernel(float* out) { out[0] = 0.0f; }

__global__ __launch_bounds__(WAVES * 32)
void cbow_loss_kernel(const int*   __restrict__ context,
                      const int*   __restrict__ target,
                      const int*   __restrict__ negatives,
                      const float* __restrict__ in_emb,
                      const float* __restrict__ out_emb,
                      float*       __restrict__ out)
{
    __shared__ __align__(16) float s_mean[WAVES][ROWS][LPAD];
    __shared__ __align__(16) float s_vec [WAVES][ROWS][LPAD];

    const int lane = threadIdx.x & 31;
    const int w    = threadIdx.x >> 5;
    const int b0   = (blockIdx.x * WAVES + w) * ROWS;
    const int d0   = lane * 4;                 // 32 lanes x float4 = 128 dims

    // ---- Phase 1: context mean (f32) for 16 rows into LDS -----------------
    for (int r = 0; r < ROWS; ++r) {
        const int b = b0 + r;
        float4 acc = make_float4(0.f, 0.f, 0.f, 0.f);
        #pragma unroll
        for (int c = 0; c < CWIN; ++c) {
            const int idx = context[b * CWIN + c];
            const float4 v = *(const float4*)(in_emb + (size_t)idx * DIM + d0);
            acc.x += v.x; acc.y += v.y; acc.z += v.z; acc.w += v.w;
        }
        const float inv = 1.0f / (float)CWIN;
        acc.x *= inv; acc.y *= inv; acc.z *= inv; acc.w *= inv;
        *(float4*)&s_mean[w][r][d0] = acc;
    }
    wave_lds_fence();

    // WMMA f32 16x16x4 operand placement: lane holds row M = lane&15,
    // K = 4*kc + 2*(lane>>4) + {0,1}. Hoist all A-fragments (mean tile is
    // invariant across the 11 score columns).
    const int mrow = lane & 15;
    const int half = lane >> 4;

    v2f afrag[DIM / 4];
    #pragma unroll
    for (int kc = 0; kc < DIM / 4; ++kc)
        afrag[kc] = *(const v2f*)&s_mean[w][mrow][4 * kc + 2 * half];

    // Diagonal of 16x16 f32 C/D: lanes 0..7 hold D[n][n] in VGPR n (n=lane);
    // lanes 24..31 hold D[n][n] in VGPR n-8 (n=lane-16).
    const bool diag_valid = (lane < 8) || (lane >= 24);
    const int  diag_r     = (lane < 8) ? lane : (lane - 24);

    float lsum = 0.0f;

    for (int j = 0; j <= KNEG; ++j) {
        // Gather the 16 output-embedding vectors for score column j.
        for (int r = 0; r < ROWS; ++r) {
            const int b   = b0 + r;
            const int idx = (j == 0) ? target[b] : negatives[b * KNEG + (j - 1)];
            *(float4*)&s_vec[w][r][d0] =
                *(const float4*)(out_emb + (size_t)idx * DIM + d0);
        }
        wave_lds_fence();

        // D[m][n] = sum_k mean[b_m][k] * vec[b_n][k]; K=128 via 32 WMMAs on
        // two independent accumulator chains.
        v8f acc0 = {}, acc1 = {};
        #pragma unroll
        for (int kc = 0; kc < DIM / 4; kc += 2) {
            const v2f bf0 = *(const v2f*)&s_vec[w][mrow][4 * kc + 2 * half];
            const v2f bf1 = *(const v2f*)&s_vec[w][mrow][4 * (kc + 1) + 2 * half];
            acc0 = __builtin_amdgcn_wmma_f32_16x16x4_f32(
                       false, afrag[kc],     false, bf0, (short)0, acc0, false, false);
            acc1 = __builtin_amdgcn_wmma_f32_16x16x4_f32(
                       false, afrag[kc + 1], false, bf1, (short)0, acc1, false, false);
        }
        const v8f acc = acc0 + acc1;

        // Extract diagonal score and accumulate log-sigmoid loss.
        float s;
        switch (diag_r) {
            case 0: s = acc[0]; break;
            case 1: s = acc[1]; break;
            case 2: s = acc[2]; break;
            case 3: s = acc[3]; break;
            case 4: s = acc[4]; break;
            case 5: s = acc[5]; break;
            case 6: s = acc[6]; break;
            default: s = acc[7]; break;
        }
        if (diag_valid) {
            const float sc  = (j == 0) ? s : -s;
            const float sig = 1.0f / (1.0f + __expf(-sc));
            lsum += __logf(sig + EPSF);
        }
        wave_lds_fence();   // drain reads of s_vec before next j overwrites it
    }

    // Wave reduction (invalid lanes carry 0), then one atomic per wave.
    #pragma unroll
    for (int off = 16; off > 0; off >>= 1)
        lsum += __shfl_down(lsum, off, 32);
    if (lane == 0)
        atomicAdd(out, -lsum * (1.0f / (float)BN));
}

extern "C" void kernel_launch(void* const* d_in, const int* in_sizes, int n_in,
                              void* d_out, int out_size, void* d_ws, size_t ws_size,
                              hipStream_t stream)
{
    (void)in_sizes; (void)n_in; (void)out_size; (void)d_ws; (void)ws_size;

    const int*   context   = (const int*)  d_in[0];
    const int*   target    = (const int*)  d_in[1];
    const int*   negatives = (const int*)  d_in[2];
    const float* in_emb    = (const float*)d_in[3];
    const float* out_emb   = (const float*)d_in[4];
    float*       out       = (float*)      d_out;

    cbow_zero_kernel<<<1, 1, 0, stream>>>(out);

    const dim3 grid(BN / (ROWS * WAVES));
    const dim3 block(WAVES * 32);
    cbow_loss_kernel<<<grid, block, 0, stream>>>(context, target, negatives,
                                                 in_emb, out_emb, out);
}